// ContrastiveCosineLoss_20083267076153
// MI455X (gfx1250) — compile-verified
//
#include <hip/hip_runtime.h>
#include <math.h>

// ---------------------------------------------------------------------------
// ContrastiveCosineLoss on MI455X (gfx1250, wave32, WMMA).
//
// loss = mean_{i<j} ( cosF(i,j) - cosR(i,j) )^2
//   cosF from full_embeddings  (2048 x 1024, f32)
//   cosR from reduced_embeddings (2048 x 128, f32)
//
// Strategy: normalize rows once, then per 32x32 Gram tile run
// V_WMMA_F32_16X16X4_F32: accumulate the reduced Gram, negate, keep
// accumulating the full Gram -> accumulator = S_full - S_red. Mask the
// upper triangle, square, reduce, atomic-add (double), finalize.
// ---------------------------------------------------------------------------

typedef __attribute__((ext_vector_type(2))) float v2f;
typedef __attribute__((ext_vector_type(8))) float v8f;

#define N_ROWS 2048
#define D_FULL 1024
#define D_RED  128
#define N_PAIRS 2096128.0   // 2048*2047/2
#define EPS 1e-8f

static __device__ __forceinline__ v8f wmma_f32(v2f a, v2f b, v8f c) {
  // 8 args: (neg_a, A, neg_b, B, c_mod, C, reuse_a, reuse_b)
  return __builtin_amdgcn_wmma_f32_16x16x4_f32(
      /*neg_a=*/false, a, /*neg_b=*/false, b,
      /*c_mod=*/(short)0, c, /*reuse_a=*/false, /*reuse_b=*/false);
}

// ---------------- row normalization (norm clamped at EPS) -------------------
template <int D, int BLK>
__global__ void normalize_rows(const float* __restrict__ x, float* __restrict__ xn) {
  __shared__ float warpsum[BLK / 32];
  const int row = blockIdx.x;
  const float* xr = x + (size_t)row * D;

  float s = 0.f;
  for (int c = threadIdx.x; c < D; c += BLK) {
    float v = xr[c];
    s += v * v;
  }
#pragma unroll
  for (int o = 16; o >= 1; o >>= 1) s += __shfl_xor(s, o, 32);

  const int wid = threadIdx.x >> 5;
  const int lane = threadIdx.x & 31;
  if (lane == 0) warpsum[wid] = s;
  __syncthreads();
  if (wid == 0) {
    float t = (lane < (BLK / 32)) ? warpsum[lane] : 0.f;
#pragma unroll
    for (int o = (BLK / 64); o >= 1; o >>= 1) t += __shfl_xor(t, o, 32);
    if (lane == 0) warpsum[0] = t;
  }
  __syncthreads();

  const float inv = 1.f / fmaxf(sqrtf(warpsum[0]), EPS);
  for (int c = threadIdx.x; c < D; c += BLK) {
    xn[(size_t)row * D + c] = xr[c] * inv;
  }
}

// ---------------- fused dual-Gram tile + masked squared-diff ----------------
// One wave (32 lanes) per 32x32 output tile. Grid (64,64); bj<bi blocks
// exit uniformly (EXEC stays all-ones around every WMMA).
//
// Fragment layout (ISA 7.12.2, 32-bit A 16x4): lane L -> M = L&15,
// VGPR0/1 hold K = 2*(L>>4) and 2*(L>>4)+1. B (4x16) mirrors with N = L&15.
// Both are plain float2 loads from row-major normalized matrices.
__global__ __launch_bounds__(32) void pair_cos_tile(
    const float* __restrict__ Xf,  // normalized full,    2048 x 1024
    const float* __restrict__ Xr,  // normalized reduced, 2048 x 128
    double* __restrict__ acc) {
  const int bi = blockIdx.y;
  const int bj = blockIdx.x;
  if (bj < bi) return;  // block-uniform: wave-wide exit

  const int lane = threadIdx.x;
  const int half = lane >> 4;   // selects K pair
  const int l16 = lane & 15;    // M (for A) / N (for B)
  const int i0 = bi * 32;
  const int j0 = bj * 32;

  v8f c00 = {}, c01 = {}, c10 = {}, c11 = {};

  // Phase 1: reduced Gram (K = 128).
  {
    const float* a0p = Xr + (size_t)(i0 + l16) * D_RED + 2 * half;
    const float* a1p = a0p + (size_t)16 * D_RED;
    const float* b0p = Xr + (size_t)(j0 + l16) * D_RED + 2 * half;
    const float* b1p = b0p + (size_t)16 * D_RED;
#pragma unroll 4
    for (int k = 0; k < D_RED; k += 4) {
      v2f a0 = *(const v2f*)(a0p + k);
      v2f a1 = *(const v2f*)(a1p + k);
      v2f b0 = *(const v2f*)(b0p + k);
      v2f b1 = *(const v2f*)(b1p + k);
      c00 = wmma_f32(a0, b0, c00);
      c01 = wmma_f32(a0, b1, c01);
      c10 = wmma_f32(a1, b0, c10);
      c11 = wmma_f32(a1, b1, c11);
    }
  }

  // Negate: accumulator becomes -S_red; full Gram accumulates on top.
  c00 = -c00; c01 = -c01; c10 = -c10; c11 = -c11;

  // Phase 2: full Gram (K = 1024) -> c = S_full - S_red.
  {
    const float* a0p = Xf + (size_t)(i0 + l16) * D_FULL + 2 * half;
    const float* a1p = a0p + (size_t)16 * D_FULL;
    const float* b0p = Xf + (size_t)(j0 + l16) * D_FULL + 2 * half;
    const float* b1p = b0p + (size_t)16 * D_FULL;
#pragma unroll 4
    for (int k = 0; k < D_FULL; k += 4) {
      v2f a0 = *(const v2f*)(a0p + k);
      v2f a1 = *(const v2f*)(a1p + k);
      v2f b0 = *(const v2f*)(b0p + k);
      v2f b1 = *(const v2f*)(b1p + k);
      c00 = wmma_f32(a0, b0, c00);
      c01 = wmma_f32(a0, b1, c01);
      c10 = wmma_f32(a1, b0, c10);
      c11 = wmma_f32(a1, b1, c11);
    }
  }

  // C/D layout: VGPR r, lane L -> M = r + 8*(L>>4), N = L&15.
  // Keep only strict upper triangle (global j > global i).
  float lsum = 0.f;
  const int giBase = i0 + 8 * half;
  const int gj = j0 + l16;
#pragma unroll
  for (int r = 0; r < 8; ++r) {
    const int giA = giBase + r;       // rows i0..i0+15 sub-tile
    const int giB = giA + 16;        // rows i0+16..i0+31 sub-tile
    float d;
    d = c00[r]; if (gj      > giA) lsum += d * d;
    d = c01[r]; if (gj + 16 > giA) lsum += d * d;
    d = c10[r]; if (gj      > giB) lsum += d * d;
    d = c11[r]; if (gj + 16 > giB) lsum += d * d;
  }

#pragma unroll
  for (int o = 16; o >= 1; o >>= 1) lsum += __shfl_xor(lsum, o, 32);
  if (lane == 0) atomicAdd(acc, (double)lsum);
}

// ---------------- tiny helpers ----------------
__global__ void zero_acc(double* acc) {
  if (threadIdx.x == 0 && blockIdx.x == 0) acc[0] = 0.0;
}

__global__ void finalize_mean(const double* __restrict__ acc, float* __restrict__ out) {
  out[0] = (float)(acc[0] / N_PAIRS);
}

// ---------------- launch ----------------
extern "C" void kernel_launch(void* const* d_in, const int* in_sizes, int n_in,
                              void* d_out, int out_size, void* d_ws, size_t ws_size,
                              hipStream_t stream) {
  const float* xr = (const float*)d_in[0];  // reduced_embeddings (2048 x 128)
  const float* xf = (const float*)d_in[1];  // full_embeddings    (2048 x 1024)

  char* ws = (char*)d_ws;
  double* acc = (double*)ws;                                    // 8 B
  float* xfn = (float*)(ws + 256);                              // 8 MB
  float* xrn = (float*)(ws + 256 + (size_t)N_ROWS * D_FULL * sizeof(float));  // 1 MB

  zero_acc<<<1, 32, 0, stream>>>(acc);
  normalize_rows<D_FULL, 256><<<N_ROWS, 256, 0, stream>>>(xf, xfn);
  normalize_rows<D_RED, 128><<<N_ROWS, 128, 0, stream>>>(xr, xrn);

  dim3 grid(N_ROWS / 32, N_ROWS / 32);  // (bj, bi); lower-triangle blocks exit
  pair_cos_tile<<<grid, 32, 0, stream>>>(xfn, xrn, acc);

  finalize_mean<<<1, 1, 0, stream>>>(acc, (float*)d_out);
}